// RandomEigenframeFilter_81372450390440
// MI455X (gfx1250) — compile-verified
//
#include <hip/hip_runtime.h>

// ---------------------------------------------------------------------------
// RandomEigenframeFilter on MI455X (gfx1250), fp32 via V_WMMA_F32_16X16X4_F32.
// Memory-bound problem: D is streamed twice (~330 MB) out of 23.3 TB/s HBM.
// ---------------------------------------------------------------------------

typedef float v2f __attribute__((ext_vector_type(2)));
typedef float v8f __attribute__((ext_vector_type(8)));

#define BATCH   8
#define MDIM    256   // KM = k_bands * m_samples
#define IDIM    64
#define ODIM    64
#define SPLIT_A 50    // split-K workgroups per batch in phase A
#define SPLIT_C 40    // workgroups per batch in phase C

__device__ __forceinline__ v8f wmma_f32(v2f a, v2f b, v8f c) {
    // 8 args: (neg_a, A, neg_b, B, c_mod, C, reuse_a, reuse_b)
    return __builtin_amdgcn_wmma_f32_16x16x4_f32(
        false, a, false, b, (short)0, c, false, false);
}

// --------------------------------------------------------------------------
__global__ __launch_bounds__(256) void zero_f32(float* __restrict__ p, int n) {
    int i = blockIdx.x * 256 + threadIdx.x;
    if (i < n) p[i] = 0.0f;
}

// --------------------------------------------------------------------------
// Phase A:  Dtx[b][m][i] += sum_n D[b][n][m] * x[b][n][i]   (split-K over n)
// A tile = D^T 16m x 4n, B tile = x 4n x 16i, C = 16m x 16i.
__global__ __launch_bounds__(256) void phaseA(const float* __restrict__ D,
                                              const float* __restrict__ X,
                                              float* __restrict__ Dtx,
                                              int N, int chunk) {
    const int b    = blockIdx.x / SPLIT_A;
    const int s    = blockIdx.x % SPLIT_A;
    const int w    = threadIdx.x >> 5;    // wave id (wave32)
    const int lane = threadIdx.x & 31;
    const int half = lane >> 4;           // lane-half selects K pair {0,1} vs {2,3}
    const int lr   = lane & 15;

    const size_t dBase = (size_t)b * N * MDIM;
    const size_t xBase = (size_t)b * N * IDIM;

    const int n_begin = s * chunk;
    const int n_end   = min(n_begin + chunk, N);

    v8f acc[2][4] = {};                   // wave owns m-tiles {2w, 2w+1} x all 4 i-tiles

    const int m0a = (2 * w)     * 16;
    const int m0b = (2 * w + 1) * 16;

    for (int k0 = n_begin; k0 + 3 < n_end; k0 += 4) {
        const int r0 = k0 + 2 * half;
        // A fragments: lanes 0-15 = m rows, VGPR0/1 = K and K+1 of this lane-half
        const float* dRow0 = D + dBase + (size_t)r0 * MDIM;
        const float* dRow1 = dRow0 + MDIM;
        v2f a0; a0.x = dRow0[m0a + lr]; a0.y = dRow1[m0a + lr];
        v2f a1; a1.x = dRow0[m0b + lr]; a1.y = dRow1[m0b + lr];
        // B fragments: lanes 0-15 = i cols, same K striping
        const float* xRow0 = X + xBase + (size_t)r0 * IDIM;
        const float* xRow1 = xRow0 + IDIM;
#pragma unroll
        for (int it = 0; it < 4; ++it) {
            v2f bf; bf.x = xRow0[it * 16 + lr]; bf.y = xRow1[it * 16 + lr];
            acc[0][it] = wmma_f32(a0, bf, acc[0][it]);
            acc[1][it] = wmma_f32(a1, bf, acc[1][it]);
        }
    }

    // C/D layout: VGPR r -> M = r (lanes 0-15) or r+8 (lanes 16-31), N = lane&15
#pragma unroll
    for (int t = 0; t < 2; ++t)
#pragma unroll
        for (int it = 0; it < 4; ++it)
#pragma unroll
            for (int r = 0; r < 8; ++r) {
                const int m = (2 * w + t) * 16 + r + half * 8;
                const int i = it * 16 + lr;
                unsafeAtomicAdd(&Dtx[((size_t)b * MDIM + m) * IDIM + i],
                                acc[t][it][r]);
            }
}

// --------------------------------------------------------------------------
// Phase B:  S[b][m][o] = (1/16) * sum_i coeff[o][i][m] * Dtx[b][m][i]
__global__ __launch_bounds__(256) void phaseB(const float* __restrict__ coeff,
                                              const float* __restrict__ Dtx,
                                              float* __restrict__ S) {
    const int g = blockIdx.x * 256 + threadIdx.x;   // [0, B*O*M)
    const int b = g / (ODIM * MDIM);
    const int r = g % (ODIM * MDIM);
    const int o = r / MDIM;
    const int m = r % MDIM;                          // lanes -> consecutive m (coalesced on coeff)
    const float* cPtr = coeff + (size_t)o * IDIM * MDIM + m;
    const float* dPtr = Dtx + ((size_t)b * MDIM + m) * IDIM;
    float sum = 0.0f;
#pragma unroll 8
    for (int i = 0; i < IDIM; ++i)
        sum += cPtr[(size_t)i * MDIM] * dPtr[i];
    S[((size_t)b * MDIM + m) * ODIM + o] = sum * 0.0625f;   // 1/sqrt(256)
}

// --------------------------------------------------------------------------
// Phase C:  out[b][n][o] = sum_m D[b][n][m] * S[b][m][o]
// S[b] (64 KB) staged in LDS; A from D via aligned float2 loads.
__global__ __launch_bounds__(256) void phaseC(const float* __restrict__ D,
                                              const float* __restrict__ S,
                                              float* __restrict__ out,
                                              int N) {
    __shared__ float s_lds[MDIM * ODIM];            // 64 KB
    const int b    = blockIdx.x / SPLIT_C;
    const int s    = blockIdx.x % SPLIT_C;
    const int w    = threadIdx.x >> 5;
    const int lane = threadIdx.x & 31;
    const int half = lane >> 4;
    const int lr   = lane & 15;

    {   // coalesced float4 fill of LDS
        const float4* src = (const float4*)(S + (size_t)b * MDIM * ODIM);
        float4* dst = (float4*)s_lds;
        for (int idx = threadIdx.x; idx < (MDIM * ODIM) / 4; idx += 256)
            dst[idx] = src[idx];
    }
    __syncthreads();

    const size_t dBase = (size_t)b * N * MDIM;
    const size_t oBase = (size_t)b * N * ODIM;
    const int ntiles = N >> 4;                      // 20000/16 = 1250

    for (int gt = s * 8 + w; gt < ntiles; gt += SPLIT_C * 8) {
        const int n0 = gt * 16;
        v8f acc[4] = {};
        const float* dRow = D + dBase + (size_t)(n0 + lr) * MDIM;
#pragma unroll 4
        for (int m0 = 0; m0 < MDIM; m0 += 4) {
            // A: lanes 0-15 = n rows; this lane-half's K pair = contiguous float2
            const float2 av = *(const float2*)(dRow + m0 + 2 * half);
            v2f a; a.x = av.x; a.y = av.y;
            // B: lanes = o cols; K from LDS rows (bank-conflict free: banks = o%64)
            const float* sRow0 = s_lds + (m0 + 2 * half) * ODIM;
            const float* sRow1 = sRow0 + ODIM;
#pragma unroll
            for (int ot = 0; ot < 4; ++ot) {
                v2f bf; bf.x = sRow0[ot * 16 + lr]; bf.y = sRow1[ot * 16 + lr];
                acc[ot] = wmma_f32(a, bf, acc[ot]);
            }
        }
#pragma unroll
        for (int ot = 0; ot < 4; ++ot)
#pragma unroll
            for (int r = 0; r < 8; ++r) {
                const int n = n0 + r + half * 8;
                out[oBase + (size_t)n * ODIM + ot * 16 + lr] = acc[ot][r];
            }
    }
}

// --------------------------------------------------------------------------
extern "C" void kernel_launch(void* const* d_in, const int* in_sizes, int n_in,
                              void* d_out, int out_size, void* d_ws, size_t ws_size,
                              hipStream_t stream) {
    const float* D     = (const float*)d_in[0];   // (B, N, M)
    const float* X     = (const float*)d_in[1];   // (B, N, I)
    const float* coeff = (const float*)d_in[2];   // (1, O, I, M)
    float* out = (float*)d_out;                   // (B, N, O)

    const int N = in_sizes[1] / (BATCH * IDIM);

    float* Dtx = (float*)d_ws;                            // B*M*I floats (512 KB)
    float* S   = Dtx + (size_t)BATCH * MDIM * IDIM;       // B*M*O floats (512 KB)

    const int nz = BATCH * MDIM * IDIM;
    hipLaunchKernelGGL(zero_f32, dim3((nz + 255) / 256), dim3(256), 0, stream,
                       Dtx, nz);

    int chunk = (N + SPLIT_A - 1) / SPLIT_A;
    chunk = (chunk + 3) & ~3;                             // multiple of 4 (K step)
    hipLaunchKernelGGL(phaseA, dim3(BATCH * SPLIT_A), dim3(256), 0, stream,
                       D, X, Dtx, N, chunk);

    hipLaunchKernelGGL(phaseB, dim3((BATCH * ODIM * MDIM) / 256), dim3(256), 0,
                       stream, coeff, Dtx, S);

    hipLaunchKernelGGL(phaseC, dim3(BATCH * SPLIT_C), dim3(256), 0, stream,
                       D, S, out, N);
}